// DiffusionLayer_69458211111592
// MI455X (gfx1250) — compile-verified
//
#include <hip/hip_runtime.h>

#define B_DIM 16
#define F_DIM 64
#define C_DIM 64
#define M_DIM 512
#define OUT_CH 32

typedef float v2f __attribute__((ext_vector_type(2)));
typedef float v8f __attribute__((ext_vector_type(8)));

// ---------------------------------------------------------------------------
// Kernel 1: s[b,c,m] = mean over F of x[b,f,c,m].  Coalesced in m.
// ---------------------------------------------------------------------------
__global__ void k_mean_f(const float* __restrict__ x, float* __restrict__ s) {
    int idx = blockIdx.x * blockDim.x + threadIdx.x;      // over B*C*M
    int m  = idx & (M_DIM - 1);
    int bc = idx >> 9;                                    // b*C + c
    int b  = bc >> 6;
    int c  = bc & 63;
    const float* xp = x + (((size_t)b * F_DIM) * C_DIM + c) * M_DIM + m;
    float acc = 0.f;
    #pragma unroll 8
    for (int f = 0; f < F_DIM; ++f)
        acc += xp[(size_t)f * C_DIM * M_DIM];
    s[idx] = acc * (1.0f / F_DIM);
}

// ---------------------------------------------------------------------------
// Kernel 2: r[b,c] = MLP(mean over M of s[b,c,:])  (Linear(1,16)->ELU->Linear(16,1))
// ---------------------------------------------------------------------------
__global__ void k_r_mlp(const float* __restrict__ s,
                        const float* __restrict__ w1, const float* __restrict__ b1,
                        const float* __restrict__ w2, const float* __restrict__ b2,
                        float* __restrict__ r) {
    int bc = blockIdx.x * blockDim.x + threadIdx.x;       // B*C = 1024
    if (bc >= B_DIM * C_DIM) return;
    const float* sp = s + (size_t)bc * M_DIM;
    float acc = 0.f;
    for (int m = 0; m < M_DIM; ++m) acc += sp[m];
    float rin = acc * (1.0f / M_DIM);
    float out = b2[0];
    #pragma unroll
    for (int j = 0; j < 16; ++j) {
        float h = w1[j] * rin + b1[j];
        h = (h > 0.f) ? h : (__expf(h) - 1.f);            // ELU (alpha=1)
        out += w2[j] * h;
    }
    r[bc] = out;
}

// ---------------------------------------------------------------------------
// Kernel 3: per-(b,m) Laplacian via V_WMMA_F32_16X16X4_F32.
// One wave per (b,m); 2 waves per block. A tile (64x64 f32) staged in LDS
// with stride-68 padding (conflict-free row AND column reads, 16B aligned).
//   As[d]  = sum_c A[c,d] * s[c]   (A-op = s replicated over rows; B-op = A rows)
//   deg[c] = sum_d A[c,d]          (A-op = ones;                 B-op = A cols)
// ---------------------------------------------------------------------------
__global__ void k_laplacian_wmma(const float* __restrict__ A,
                                 const float* __restrict__ s,
                                 const float* __restrict__ r,
                                 const float* __restrict__ phys,
                                 const float* __restrict__ kappa_p,
                                 const float* __restrict__ alpha_p,
                                 float* __restrict__ s_new) {
    __shared__ __align__(16) float ldsA[2][64 * 68];      // 34816 B
    __shared__ float ldsS[2][64];

    const int tid  = threadIdx.x;
    const int w    = tid >> 5;                            // wave in block (0..1)
    const int lane = tid & 31;
    const int wid  = blockIdx.x * 2 + w;                  // global (b*M + m)
    const int b    = wid >> 9;
    const int m    = wid & (M_DIM - 1);
    const int half = lane >> 4;                           // 0 or 1
    const int l15  = lane & 15;

    // ---- stage s[b, :, m] (64 values, strided by M) ----
    const float* sb = s + ((size_t)b * C_DIM) * M_DIM + m;
    ldsS[w][lane]      = sb[(size_t)lane * M_DIM];
    ldsS[w][lane + 32] = sb[(size_t)(lane + 32) * M_DIM];

    // ---- stage A[b,m] tile (4096 floats) coalesced -> padded LDS ----
    const float4* Ag4 = (const float4*)(A + (size_t)wid * 4096);
    #pragma unroll 4
    for (int i = 0; i < 32; ++i) {
        int i4 = lane + i * 32;                           // float4 index 0..1023
        float4 v = Ag4[i4];
        int row = i4 >> 4;                                // 16 float4 per row
        int col = (i4 & 15) * 4;
        *(float4*)&ldsA[w][row * 68 + col] = v;
    }
    __syncthreads();

    // ---- WMMA accumulation ----
    v8f as0 = {}, as1 = {}, as2 = {}, as3 = {};           // As per d-block
    v8f dg0 = {}, dg1 = {}, dg2 = {}, dg3 = {};           // deg per c-block
    v2f ones; ones[0] = 1.0f; ones[1] = 1.0f;

    for (int kc = 0; kc < 16; ++kc) {
        const int k0 = kc * 4 + half * 2;                 // K slot, VGPR0 (doc layout)
        const int k1 = k0 + 1;                            // K slot, VGPR1

        // A-operand for As: s replicated across all 16 rows
        v2f a_s; a_s[0] = ldsS[w][k0]; a_s[1] = ldsS[w][k1];

        // B-operands for As: rows k0/k1 of A, column chunks of 16
        const int rb0 = k0 * 68 + l15;
        const int rb1 = k1 * 68 + l15;
        v2f bb0 = { ldsA[w][rb0],      ldsA[w][rb1]      };
        v2f bb1 = { ldsA[w][rb0 + 16], ldsA[w][rb1 + 16] };
        v2f bb2 = { ldsA[w][rb0 + 32], ldsA[w][rb1 + 32] };
        v2f bb3 = { ldsA[w][rb0 + 48], ldsA[w][rb1 + 48] };
        as0 = __builtin_amdgcn_wmma_f32_16x16x4_f32(false, a_s, false, bb0, (short)0, as0, false, false);
        as1 = __builtin_amdgcn_wmma_f32_16x16x4_f32(false, a_s, false, bb1, (short)0, as1, false, false);
        as2 = __builtin_amdgcn_wmma_f32_16x16x4_f32(false, a_s, false, bb2, (short)0, as2, false, false);
        as3 = __builtin_amdgcn_wmma_f32_16x16x4_f32(false, a_s, false, bb3, (short)0, as3, false, false);

        // B-operands for deg: B[k][q] = A[cb*16+q, k]  (column reads, padded)
        const int cB = l15 * 68;
        v2f dd0 = { ldsA[w][cB + k0],             ldsA[w][cB + k1]             };
        v2f dd1 = { ldsA[w][cB + 16 * 68 + k0],   ldsA[w][cB + 16 * 68 + k1]   };
        v2f dd2 = { ldsA[w][cB + 32 * 68 + k0],   ldsA[w][cB + 32 * 68 + k1]   };
        v2f dd3 = { ldsA[w][cB + 48 * 68 + k0],   ldsA[w][cB + 48 * 68 + k1]   };
        dg0 = __builtin_amdgcn_wmma_f32_16x16x4_f32(false, ones, false, dd0, (short)0, dg0, false, false);
        dg1 = __builtin_amdgcn_wmma_f32_16x16x4_f32(false, ones, false, dd1, (short)0, dg1, false, false);
        dg2 = __builtin_amdgcn_wmma_f32_16x16x4_f32(false, ones, false, dd2, (short)0, dg2, false, false);
        dg3 = __builtin_amdgcn_wmma_f32_16x16x4_f32(false, ones, false, dd3, (short)0, dg3, false, false);
    }

    // Every lane holds As/deg for N = l15 in every block (results replicated
    // over rows).  Lanes 0-15 own c-blocks {0,1}; lanes 16-31 own {2,3}.
    const float kap = kappa_p[0];
    const float kf  = log1pf(__expf(kap));                // softplus(kappa)
    const float al  = alpha_p[0];

    float degA = half ? dg2[0] : dg0[0];
    float degB = half ? dg3[0] : dg1[0];
    float asA  = half ? as2[0] : as0[0];
    float asB  = half ? as3[0] : as1[0];

    const int   c0   = half * 32 + l15;
    const int   c1   = c0 + 16;
    const size_t obase = ((size_t)b * C_DIM) * M_DIM + m;

    {
        float sv = ldsS[w][c0];
        float ls = degA * sv - asA;
        float sn = sv - kf * ls + (r[b * C_DIM + c0] + al * phys[obase + (size_t)c0 * M_DIM]);
        s_new[obase + (size_t)c0 * M_DIM] = sn;
    }
    {
        float sv = ldsS[w][c1];
        float ls = degB * sv - asB;
        float sn = sv - kf * ls + (r[b * C_DIM + c1] + al * phys[obase + (size_t)c1 * M_DIM]);
        s_new[obase + (size_t)c1 * M_DIM] = sn;
    }
}

// ---------------------------------------------------------------------------
// Kernel 4: out[b,o,c,m] = s_new[b,c,m]*pw[o] + pb[o].  Fully coalesced 67 MB.
// ---------------------------------------------------------------------------
__global__ void k_expand(const float* __restrict__ s_new,
                         const float* __restrict__ pw, const float* __restrict__ pb,
                         float* __restrict__ out) {
    size_t idx = (size_t)blockIdx.x * blockDim.x + threadIdx.x;  // B*OUT*C*M
    int m = (int)(idx & (M_DIM - 1));
    size_t t = idx >> 9;
    int c = (int)(t & 63); t >>= 6;
    int o = (int)(t & 31);
    int b = (int)(t >> 5);
    float v = s_new[(((size_t)b * C_DIM) + c) * M_DIM + m];
    out[idx] = v * pw[o] + pb[o];
}

// ---------------------------------------------------------------------------
extern "C" void kernel_launch(void* const* d_in, const int* in_sizes, int n_in,
                              void* d_out, int out_size, void* d_ws, size_t ws_size,
                              hipStream_t stream) {
    const float* x     = (const float*)d_in[0];
    const float* A     = (const float*)d_in[1];
    const float* phys  = (const float*)d_in[2];
    const float* kappa = (const float*)d_in[3];
    const float* alpha = (const float*)d_in[4];
    const float* w1    = (const float*)d_in[5];
    const float* b1    = (const float*)d_in[6];
    const float* w2    = (const float*)d_in[7];
    const float* b2    = (const float*)d_in[8];
    const float* pw    = (const float*)d_in[9];
    const float* pb    = (const float*)d_in[10];
    float* out = (float*)d_out;

    float* ws    = (float*)d_ws;
    float* s     = ws;                                    // 524288 floats
    float* s_new = ws + (size_t)B_DIM * C_DIM * M_DIM;    // 524288 floats
    float* r     = s_new + (size_t)B_DIM * C_DIM * M_DIM; // 1024 floats

    const int nS = B_DIM * C_DIM * M_DIM;                 // 524288
    k_mean_f<<<nS / 256, 256, 0, stream>>>(x, s);
    k_r_mlp<<<(B_DIM * C_DIM + 255) / 256, 256, 0, stream>>>(s, w1, b1, w2, b2, r);
    // one wave32 per (b,m); 2 waves per block -> 4096 blocks of 64 threads
    k_laplacian_wmma<<<(B_DIM * M_DIM) / 2, 64, 0, stream>>>(A, s, r, phys, kappa, alpha, s_new);
    const size_t nOut = (size_t)B_DIM * OUT_CH * C_DIM * M_DIM; // 16777216
    k_expand<<<(unsigned)(nOut / 256), 256, 0, stream>>>(s_new, pw, pb, out);
}